// HungarianMatcher_1657857376674
// MI455X (gfx1250) — compile-verified
//
#include <hip/hip_runtime.h>
#include <hip/hip_bf16.h>
#include <math.h>
#include <stdint.h>

typedef __attribute__((ext_vector_type(2))) float v2f;
typedef __attribute__((ext_vector_type(8))) float v8f;

#define BS    64
#define NQ    100
#define TT    32
#define NROWS (BS * NQ)   // 6400
#define MCOLS (BS * TT)   // 2048
#define C1    92
#define CA    118
#define CP1   96          // padded class dim (p1/p2)
#define CPA   120         // padded class dim (action)

// ---------------------------------------------------------------------------
// Async global->LDS copy (CDNA5 GLOBAL_LOAD_ASYNC_TO_LDS_B128, GVS mode).
// lds_off : wave-relative LDS byte address (low 32 bits of a shared pointer)
// base    : 64-bit global base (SGPR pair), voff: per-lane byte offset
// ---------------------------------------------------------------------------
__device__ __forceinline__ void async_b128(uint32_t lds_off, uint32_t voff,
                                           uint64_t base) {
    asm volatile("global_load_async_to_lds_b128 %0, %1, %2"
                 :: "v"(lds_off), "v"(voff), "s"(base)
                 : "memory");
}
__device__ __forceinline__ void wait_async0() {
    asm volatile("s_wait_asynccnt 0x0" ::: "memory");
}
__device__ __forceinline__ uint32_t lds_addr32(const void* p) {
    return (uint32_t)(uintptr_t)p;   // LDS aperture: addr[31:0] == LDS offset
}

// ---------------------------------------------------------------------------
// Kernel 1: softmax per row for the three heads, zero-padded to CP1/CPA cols.
// One wave (32 lanes) per row. ws layout (floats):
//   [0 .. 6400*96)            p1_prob
//   [6400*96 .. 2*6400*96)    p2_prob
//   [2*6400*96 .. +6400*120)  act_prob
// ---------------------------------------------------------------------------
__global__ __launch_bounds__(32)
void softmax_pad_kernel(const float* __restrict__ p1_logits,
                        const float* __restrict__ p2_logits,
                        const float* __restrict__ act_logits,
                        float* __restrict__ ws) {
    const int row  = blockIdx.x;   // 0..6399
    const int head = blockIdx.y;   // 0..2
    const int lane = threadIdx.x;  // 0..31

    const float* src;
    float* dst;
    int C, CPd;
    if (head == 0) {
        src = p1_logits + (size_t)row * C1;
        dst = ws + (size_t)row * CP1;
        C = C1; CPd = CP1;
    } else if (head == 1) {
        src = p2_logits + (size_t)row * C1;
        dst = ws + (size_t)NROWS * CP1 + (size_t)row * CP1;
        C = C1; CPd = CP1;
    } else {
        src = act_logits + (size_t)row * CA;
        dst = ws + (size_t)2 * NROWS * CP1 + (size_t)row * CPA;
        C = CA; CPd = CPA;
    }

    float x[4];
    float mx = -INFINITY;
    int cnt = 0;
    for (int c = lane; c < C; c += 32) { x[cnt] = src[c]; mx = fmaxf(mx, x[cnt]); cnt++; }
    for (int o = 16; o > 0; o >>= 1) mx = fmaxf(mx, __shfl_xor(mx, o, 32));
    float s = 0.f;
    for (int k = 0; k < cnt; k++) { x[k] = expf(x[k] - mx); s += x[k]; }
    for (int o = 16; o > 0; o >>= 1) s += __shfl_xor(s, o, 32);
    const float inv = 1.0f / s;
    cnt = 0;
    for (int c = lane; c < CPd; c += 32) {
        float vv = 0.0f;
        if (c < C) { vv = x[cnt] * inv; cnt++; }
        dst[c] = vv;
    }
}

// ---------------------------------------------------------------------------
// Box cost: COST_BBOX * L1(cxcywh) - COST_GIOU * GIoU(xyxy)
// ---------------------------------------------------------------------------
__device__ __forceinline__ float box_cost(const float4 p, const float4 t) {
    const float l1 = fabsf(p.x - t.x) + fabsf(p.y - t.y) +
                     fabsf(p.z - t.z) + fabsf(p.w - t.w);
    const float px1 = p.x - 0.5f * p.z, py1 = p.y - 0.5f * p.w;
    const float px2 = p.x + 0.5f * p.z, py2 = p.y + 0.5f * p.w;
    const float tx1 = t.x - 0.5f * t.z, ty1 = t.y - 0.5f * t.w;
    const float tx2 = t.x + 0.5f * t.z, ty2 = t.y + 0.5f * t.w;
    const float areaP = (px2 - px1) * (py2 - py1);
    const float areaT = (tx2 - tx1) * (ty2 - ty1);
    const float iw = fmaxf(fminf(px2, tx2) - fmaxf(px1, tx1), 0.0f);
    const float ih = fmaxf(fminf(py2, ty2) - fmaxf(py1, ty1), 0.0f);
    const float inter = iw * ih;
    const float uni = areaP + areaT - inter;
    const float iou = inter / uni;
    const float cw = fmaxf(fmaxf(px2, tx2) - fminf(px1, tx1), 0.0f);
    const float ch = fmaxf(fmaxf(py2, ty2) - fminf(py1, ty1), 0.0f);
    const float areaC = cw * ch;
    const float giou = iou - (areaC - uni) / areaC;
    return l1 - giou;
}

// ---------------------------------------------------------------------------
// Kernel 2: cost matrix build. Block = 128 threads (4 waves); block tile =
// 16 rows x 64 cols; each wave owns a 16x16 tile. Class cost is an exact
// one-hot fp32 GEMM on V_WMMA_F32_16X16X4_F32 (weights folded into B):
//   acc = sum_h  P_h[16 x CPh] * (-w_h * OneHot_h[CPh x 16])
// with w = {0.3, 0.3, 0.6}  (= BETA1 * COST_CLASS * alpha_h / 4).
// Prob tiles + pred boxes staged via GLOBAL_LOAD_ASYNC_TO_LDS_B128.
// Box cost (BETA2 * (cbox_h + cbox_o)/2) added with VALU, single store.
// ---------------------------------------------------------------------------
__global__ __launch_bounds__(128)
void cost_kernel(const float* __restrict__ p1_boxes,
                 const float* __restrict__ p2_boxes,
                 const int*   __restrict__ p1_ids,
                 const int*   __restrict__ p2_ids,
                 const int*   __restrict__ act_ids,
                 const float* __restrict__ p1_tbox,
                 const float* __restrict__ p2_tbox,
                 const float* __restrict__ ws,
                 float* __restrict__ out) {
    __shared__ __align__(16) float s_p1[16 * CP1];
    __shared__ __align__(16) float s_p2[16 * CP1];
    __shared__ __align__(16) float s_ac[16 * CPA];
    __shared__ __align__(16) float s_pb1[16 * 4];
    __shared__ __align__(16) float s_pb2[16 * 4];

    const int R0  = blockIdx.y * 16;   // row base (pred index)
    const int CB0 = blockIdx.x * 64;   // col base (target index)
    const int tid  = threadIdx.x;
    const int wave = tid >> 5;
    const int lane = tid & 31;

    // async-stage 16 rows of padded probs + pred boxes in LDS (shared by 4 waves)
    {
        const uint64_t g1 = (uint64_t)(uintptr_t)(ws + (size_t)R0 * CP1);
        const uint64_t g2 = (uint64_t)(uintptr_t)(ws + (size_t)NROWS * CP1 + (size_t)R0 * CP1);
        const uint64_t ga = (uint64_t)(uintptr_t)(ws + (size_t)2 * NROWS * CP1 + (size_t)R0 * CPA);
        for (int i = tid; i < 16 * CP1 / 4; i += 128) {        // 384 x b128
            const uint32_t off = (uint32_t)i * 16u;
            async_b128(lds_addr32(&s_p1[i * 4]), off, g1);
            async_b128(lds_addr32(&s_p2[i * 4]), off, g2);
        }
        for (int i = tid; i < 16 * CPA / 4; i += 128) {        // 480 x b128
            const uint32_t off = (uint32_t)i * 16u;
            async_b128(lds_addr32(&s_ac[i * 4]), off, ga);
        }
        if (tid < 16) {
            const uint32_t off = (uint32_t)tid * 16u;
            async_b128(lds_addr32(&s_pb1[tid * 4]), off,
                       (uint64_t)(uintptr_t)(p1_boxes + (size_t)R0 * 4));
            async_b128(lds_addr32(&s_pb2[tid * 4]), off,
                       (uint64_t)(uintptr_t)(p2_boxes + (size_t)R0 * 4));
        }
        wait_async0();
    }
    __syncthreads();

    // WMMA fragment coordinates (ISA 7.12.2, 32-bit A 16x4 / C 16x16)
    const int n     = lane & 15;            // tile column
    const int m     = lane & 15;            // A-tile row
    const int khalf = (lane >> 4) << 1;     // K offset {0,2}
    const int jg    = CB0 + wave * 16 + n;  // global target column

    const int id1 = p1_ids[jg];
    const int id2 = p2_ids[jg];
    const int ida = act_ids[jg];

    v8f acc = {0.f, 0.f, 0.f, 0.f, 0.f, 0.f, 0.f, 0.f};

    #pragma unroll 4
    for (int k0 = 0; k0 < CP1; k0 += 4) {   // head h (weight 0.3)
        v2f a, b;
        a.x = s_p1[m * CP1 + k0 + khalf];
        a.y = s_p1[m * CP1 + k0 + khalf + 1];
        b.x = (id1 == k0 + khalf)     ? -0.3f : 0.0f;
        b.y = (id1 == k0 + khalf + 1) ? -0.3f : 0.0f;
        acc = __builtin_amdgcn_wmma_f32_16x16x4_f32(false, a, false, b,
                                                    (short)0, acc, false, false);
    }
    #pragma unroll 4
    for (int k0 = 0; k0 < CP1; k0 += 4) {   // head o (weight 0.3)
        v2f a, b;
        a.x = s_p2[m * CP1 + k0 + khalf];
        a.y = s_p2[m * CP1 + k0 + khalf + 1];
        b.x = (id2 == k0 + khalf)     ? -0.3f : 0.0f;
        b.y = (id2 == k0 + khalf + 1) ? -0.3f : 0.0f;
        acc = __builtin_amdgcn_wmma_f32_16x16x4_f32(false, a, false, b,
                                                    (short)0, acc, false, false);
    }
    #pragma unroll 4
    for (int k0 = 0; k0 < CPA; k0 += 4) {   // head r (weight 0.6)
        v2f a, b;
        a.x = s_ac[m * CPA + k0 + khalf];
        a.y = s_ac[m * CPA + k0 + khalf + 1];
        b.x = (ida == k0 + khalf)     ? -0.6f : 0.0f;
        b.y = (ida == k0 + khalf + 1) ? -0.6f : 0.0f;
        acc = __builtin_amdgcn_wmma_f32_16x16x4_f32(false, a, false, b,
                                                    (short)0, acc, false, false);
    }

    // box costs + store; D layout: VGPR r <-> rows r (lanes 0-15) / r+8 (16-31)
    const float4 t1 = reinterpret_cast<const float4*>(p1_tbox)[jg];
    const float4 t2 = reinterpret_cast<const float4*>(p2_tbox)[jg];
    const float4* pb1 = reinterpret_cast<const float4*>(s_pb1);
    const float4* pb2 = reinterpret_cast<const float4*>(s_pb2);
    const int mbase = (lane >> 4) << 3;     // 0 or 8

    #pragma unroll
    for (int r = 0; r < 8; r++) {
        const int mm = r + mbase;
        const float cb1 = box_cost(pb1[mm], t1);
        const float cb2 = box_cost(pb2[mm], t2);
        const float cost = acc[r] + 0.5f * (cb1 + cb2);   // BETA2 == 1
        out[(size_t)(R0 + mm) * MCOLS + jg] = cost;
    }
}

// ---------------------------------------------------------------------------
// Kernel 3: Jonker-Volgenant LAP per batch. One wave per batch, lanes own
// columns j = lane + 32k (matches numpy argmin first-index tie-breaking).
// Cost transposed to [32 targets x 100 queries], double arithmetic.
// ---------------------------------------------------------------------------
__global__ __launch_bounds__(32)
void lap_kernel(const float* __restrict__ Cfull, int* __restrict__ outIdx) {
    const int b    = blockIdx.x;
    const int lane = threadIdx.x;

    __shared__ float  Dc[TT * NQ];
    __shared__ double u[TT], v[NQ], shortest[NQ];
    __shared__ int    path[NQ], row4col[NQ], col4row[TT];
    __shared__ unsigned char SC[NQ], SR[TT];
    __shared__ int    sh_i, sh_sink;
    __shared__ double sh_min;

    const float* Cb = Cfull + (size_t)b * NQ * MCOLS + (size_t)b * TT;
    for (int idx = lane; idx < TT * NQ; idx += 32) {
        const int i = idx / NQ;           // target (row of D)
        const int j = idx - i * NQ;       // query  (col of D)
        Dc[idx] = Cb[(size_t)j * MCOLS + i];
    }
    for (int j = lane; j < NQ; j += 32) { v[j] = 0.0; row4col[j] = -1; path[j] = -1; }
    u[lane] = 0.0;
    col4row[lane] = -1;
    __syncthreads();

    for (int cur = 0; cur < TT; cur++) {
        for (int j = lane; j < NQ; j += 32) { shortest[j] = INFINITY; SC[j] = 0; }
        SR[lane] = 0;
        if (lane == 0) { sh_i = cur; sh_sink = -1; sh_min = 0.0; }
        __syncthreads();

        while (true) {
            const int    i      = sh_i;
            const double minVal = sh_min;
            if (lane == 0) SR[i] = 1;
            const double ui = u[i];

            double bestv = INFINITY;
            int    bestj = -1;
            for (int j = lane; j < NQ; j += 32) {
                if (!SC[j]) {
                    const double r = minVal + (double)Dc[i * NQ + j] - ui - v[j];
                    if (r < shortest[j]) { shortest[j] = r; path[j] = i; }
                    const double sj = shortest[j];
                    if (sj < bestv) { bestv = sj; bestj = j; }
                }
            }
            // wave-wide argmin: min value, tie -> smallest column index
            for (int o = 16; o > 0; o >>= 1) {
                const double ov = __shfl_xor(bestv, o, 32);
                const int    oj = __shfl_xor(bestj, o, 32);
                if (ov < bestv || (ov == bestv && (unsigned)oj < (unsigned)bestj)) {
                    bestv = ov; bestj = oj;
                }
            }
            __syncthreads();                 // shortest[] writes visible
            if (lane == 0) {
                const double mv = shortest[bestj];
                SC[bestj] = 1;
                sh_min = mv;
                const int r4 = row4col[bestj];
                if (r4 == -1) sh_sink = bestj; else sh_i = r4;
            }
            __syncthreads();
            if (sh_sink != -1) break;
        }

        const double minVal = sh_min;
        const int    sink   = sh_sink;
        // dual updates (pre-augmentation col4row, as in reference)
        if (lane == cur)      u[lane] += minVal;
        else if (SR[lane])    u[lane] += minVal - shortest[col4row[lane]];
        for (int j = lane; j < NQ; j += 32)
            if (SC[j]) v[j] -= minVal - shortest[j];
        __syncthreads();
        if (lane == 0) {                     // augment along path
            int j = sink;
            while (true) {
                const int i = path[j];
                row4col[j] = i;
                const int tmp = col4row[i];
                col4row[i] = j;
                j = tmp;
                if (i == cur) break;
            }
        }
        __syncthreads();
    }

    // order = argsort(col4row); row_ind = col4row[order]; col_ind = order
    const int cj = col4row[lane];
    int rank = 0;
    for (int k = 0; k < TT; k++) rank += (col4row[k] < cj) ? 1 : 0;
    int* base = outIdx + (size_t)b * 2 * TT;
    base[rank]      = cj;    // query index
    base[TT + rank] = lane;  // target index
}

// ---------------------------------------------------------------------------
extern "C" void kernel_launch(void* const* d_in, const int* in_sizes, int n_in,
                              void* d_out, int out_size, void* d_ws, size_t ws_size,
                              hipStream_t stream) {
    const float* p1_logits  = (const float*)d_in[0];
    const float* p1_boxes   = (const float*)d_in[1];
    const float* p2_logits  = (const float*)d_in[2];
    const float* p2_boxes   = (const float*)d_in[3];
    const float* act_logits = (const float*)d_in[4];
    const int*   p1_ids     = (const int*)d_in[5];
    const float* p1_tbox    = (const float*)d_in[6];
    const int*   p2_ids     = (const int*)d_in[7];
    const float* p2_tbox    = (const float*)d_in[8];
    const int*   act_ids    = (const int*)d_in[9];

    float* out = (float*)d_out;
    float* ws  = (float*)d_ws;
    int*   idx = (int*)d_out + (size_t)NROWS * MCOLS;   // int32 bits after C

    dim3 gs(NROWS, 3);
    softmax_pad_kernel<<<gs, 32, 0, stream>>>(p1_logits, p2_logits, act_logits, ws);

    dim3 gc(MCOLS / 64, NROWS / 16);
    cost_kernel<<<gc, 128, 0, stream>>>(p1_boxes, p2_boxes, p1_ids, p2_ids, act_ids,
                                        p1_tbox, p2_tbox, ws, out);

    lap_kernel<<<BS, 32, 0, stream>>>(out, idx);
}